// ProjectionModel_15676630630938
// MI455X (gfx1250) — compile-verified
//
#include <hip/hip_runtime.h>
#include <hip/hip_bf16.h>

typedef float v2f __attribute__((ext_vector_type(2)));
typedef float v8f __attribute__((ext_vector_type(8)));

// ---------------------------------------------------------------------------
// Power iteration: eta = 1.8 / ||A||_2^2   (single block, 1024 threads)
// ---------------------------------------------------------------------------
__global__ void power_iter_kernel(const float* __restrict__ A, float* etaOut,
                                  int m, int d) {
  __shared__ float vv[1024];
  __shared__ float ww[1024];
  __shared__ float red[1024];
  const int t = threadIdx.x;

  vv[t] = rsqrtf((float)d);
  __syncthreads();

  for (int it = 0; it < 10; ++it) {
    // w = A v   (thread t -> row t)
    float s = 0.f;
    const float* row = A + (size_t)t * d;
    for (int k = 0; k < d; k += 4) {
      float4 a4 = *(const float4*)(row + k);
      s += a4.x * vv[k] + a4.y * vv[k + 1] + a4.z * vv[k + 2] + a4.w * vv[k + 3];
    }
    ww[t] = s;
    __syncthreads();
    // u = A^T w  (thread t -> column t)
    float u = 0.f;
    for (int i = 0; i < m; ++i) u += A[(size_t)i * d + t] * ww[i];
    red[t] = u * u;
    __syncthreads();
    for (int sft = 512; sft > 0; sft >>= 1) {
      if (t < sft) red[t] += red[t + sft];
      __syncthreads();
    }
    float nrm = sqrtf(red[0]) + 1e-12f;
    __syncthreads();
    vv[t] = u / nrm;
    __syncthreads();
  }

  // sigma2 = || A v ||^2
  float s = 0.f;
  const float* row = A + (size_t)t * d;
  for (int k = 0; k < d; k += 4) {
    float4 a4 = *(const float4*)(row + k);
    s += a4.x * vv[k] + a4.y * vv[k + 1] + a4.z * vv[k + 2] + a4.w * vv[k + 3];
  }
  red[t] = s * s;
  __syncthreads();
  for (int sft = 512; sft > 0; sft >>= 1) {
    if (t < sft) red[t] += red[t + sft];
    __syncthreads();
  }
  if (t == 0) etaOut[0] = 1.8f / red[0];
}

// ---------------------------------------------------------------------------
// AT[j][i] = A[i][j]   (A: m x d, AT: d x m)
// ---------------------------------------------------------------------------
__global__ void transpose_kernel(const float* __restrict__ A,
                                 float* __restrict__ AT, int m, int d) {
  __shared__ float tile[32][33];
  int bx = blockIdx.x * 32;  // column block of A
  int by = blockIdx.y * 32;  // row block of A
  int tx = threadIdx.x, ty = threadIdx.y;  // 32 x 8
  for (int i = 0; i < 32; i += 8)
    tile[ty + i][tx] = A[(size_t)(by + ty + i) * d + bx + tx];
  __syncthreads();
  for (int i = 0; i < 32; i += 8)
    AT[(size_t)(bx + ty + i) * m + by + tx] = tile[tx][ty + i];
}

// ---------------------------------------------------------------------------
// lam1 = relu(eta * (c - b))      elementwise over [B, m]
// ---------------------------------------------------------------------------
__global__ void init_lam_kernel(const float* __restrict__ c,
                                const float* __restrict__ b,
                                const float* __restrict__ etaPtr,
                                float* __restrict__ lam, int m, long long n) {
  long long i = (long long)blockIdx.x * blockDim.x + threadIdx.x;
  if (i >= n) return;
  float eta = etaPtr[0];
  int col = (int)(i % m);
  lam[i] = fmaxf(eta * (c[i] - b[col]), 0.f);
}

// ---------------------------------------------------------------------------
// WMMA fp32 GEMM core:  S = P * Q^T
//   P: [Mrows, K] row-major (ldP),  Q: [Ncols, K] row-major (ldQ)
//   block tile 128x128, 8 waves (256 threads), wave tile 32x64 (2x4 of 16x16)
// MODE 0: Out = S
// MODE 1: Out = relu(P_tile + eta*(aux0 - bVec - S))      (lam update; P==lamIn)
// MODE 2: Out = aux0 - S                                  (final: x - lam@A)
// ---------------------------------------------------------------------------
#define KT   64
#define LDT  68   // 68 % 64 == 4 -> conflict-free b64 fragment reads

template <int MODE>
__global__ __launch_bounds__(256)
void wmma_gemm_nt(const float* __restrict__ P, const float* __restrict__ Q,
                  float* __restrict__ Out, const float* aux0, const float* bVec,
                  const float* etaPtr, int K, int ldP, int ldQ, int ldOut) {
  __shared__ float lp[128 * LDT];
  __shared__ float lq[128 * LDT];

  const int t = threadIdx.x;
  const int lane = t & 31;
  const int wave = t >> 5;
  const int wr = wave >> 1;          // wave row (0..3), 32 rows each
  const int wc = wave & 1;           // wave col (0..1), 64 cols each
  const int fr = lane & 15;          // fragment row/col within 16
  const int fh = (lane >> 4) << 1;   // k sub-offset: 0 or 2

  const int rowBase = blockIdx.y * 128;  // output rows from P
  const int colBase = blockIdx.x * 128;  // output cols from Q rows

  // tile-loader mapping: 256 threads, float4 each -> 16 rows / pass, 8 passes
  const int lr = t >> 4;         // 0..15
  const int lc4 = (t & 15) * 4;  // 0,4,...,60

  const v8f zero = {0.f, 0.f, 0.f, 0.f, 0.f, 0.f, 0.f, 0.f};
  v8f acc[2][4];
#pragma unroll
  for (int i = 0; i < 2; ++i)
#pragma unroll
    for (int j = 0; j < 4; ++j) acc[i][j] = zero;

  for (int k0 = 0; k0 < K; k0 += KT) {
    __syncthreads();
#pragma unroll
    for (int pass = 0; pass < 8; ++pass) {
      int row = pass * 16 + lr;
      float4 vP = *(const float4*)(P + (size_t)(rowBase + row) * ldP + k0 + lc4);
      float4 vQ = *(const float4*)(Q + (size_t)(colBase + row) * ldQ + k0 + lc4);
      *(float4*)(&lp[row * LDT + lc4]) = vP;
      *(float4*)(&lq[row * LDT + lc4]) = vQ;
    }
    __syncthreads();

    // warm the caches for the next K-tile (global_prefetch_b8), zero VGPR cost
    if (k0 + KT < K) {
      int prow = t >> 1;              // 0..127
      int phalf = (t & 1) * 32;       // two 128B halves per 256B row
      __builtin_prefetch(P + (size_t)(rowBase + prow) * ldP + k0 + KT + phalf, 0, 0);
      __builtin_prefetch(Q + (size_t)(colBase + prow) * ldQ + k0 + KT + phalf, 0, 0);
    }

#pragma unroll
    for (int kk = 0; kk < KT; kk += 4) {
      v2f a[2], bf[4];
#pragma unroll
      for (int mt = 0; mt < 2; ++mt)
        a[mt] = *(const v2f*)(&lp[(wr * 32 + mt * 16 + fr) * LDT + kk + fh]);
#pragma unroll
      for (int nt = 0; nt < 4; ++nt)
        bf[nt] = *(const v2f*)(&lq[(wc * 64 + nt * 16 + fr) * LDT + kk + fh]);
#pragma unroll
      for (int mt = 0; mt < 2; ++mt)
#pragma unroll
        for (int nt = 0; nt < 4; ++nt)
          acc[mt][nt] = __builtin_amdgcn_wmma_f32_16x16x4_f32(
              false, a[mt], false, bf[nt], (short)0, acc[mt][nt], false, false);
    }
  }

  // -------- epilogue --------
  const float eta = (MODE == 1) ? etaPtr[0] : 0.f;
  const int cLane = lane & 15;
  const int rHalf = (lane >> 4) * 8;

#pragma unroll
  for (int mt = 0; mt < 2; ++mt) {
#pragma unroll
    for (int nt = 0; nt < 4; ++nt) {
#pragma unroll
      for (int g = 0; g < 8; ++g) {
        int row = rowBase + wr * 32 + mt * 16 + g + rHalf;
        int col = colBase + wc * 64 + nt * 16 + cLane;
        size_t idx = (size_t)row * ldOut + col;
        float s = acc[mt][nt][g];
        if (MODE == 0) {
          Out[idx] = s;
        } else if (MODE == 1) {
          float lamv = P[(size_t)row * ldP + col];
          Out[idx] = fmaxf(lamv + eta * (aux0[idx] - bVec[col] - s), 0.f);
        } else {
          Out[idx] = aux0[idx] - s;
        }
      }
    }
  }
}

// ---------------------------------------------------------------------------
extern "C" void kernel_launch(void* const* d_in, const int* in_sizes, int n_in,
                              void* d_out, int out_size, void* d_ws,
                              size_t ws_size, hipStream_t stream) {
  const float* x = (const float*)d_in[0];
  const float* A = (const float*)d_in[1];
  const float* b = (const float*)d_in[2];
  float* out = (float*)d_out;

  const int m = in_sizes[2];
  const int d = in_sizes[1] / m;
  const int B = in_sizes[0] / d;

  // workspace layout (floats)
  float* ws = (float*)d_ws;
  float* eta  = ws;                       // 64 floats padding
  float* G    = eta + 64;                 // m*m
  float* AT   = G + (size_t)m * m;        // d*m
  float* cMat = AT + (size_t)d * m;       // B*m
  float* lamA = cMat + (size_t)B * m;     // B*m
  float* lamB = lamA + (size_t)B * m;     // B*m

  // 1) eta = 1.8 / sigma_max^2
  power_iter_kernel<<<1, 1024, 0, stream>>>(A, eta, m, d);

  // 2) AT = A^T
  {
    dim3 grid(d / 32, m / 32), blk(32, 8);
    transpose_kernel<<<grid, blk, 0, stream>>>(A, AT, m, d);
  }

  // 3) G = A A^T   [m x m]
  wmma_gemm_nt<0><<<dim3(m / 128, m / 128), 256, 0, stream>>>(
      A, A, G, nullptr, nullptr, nullptr, d, d, d, m);

  // 4) c = x A^T   [B x m]
  wmma_gemm_nt<0><<<dim3(m / 128, B / 128), 256, 0, stream>>>(
      x, A, cMat, nullptr, nullptr, nullptr, d, d, d, m);

  // 5) lam1 = relu(eta * (c - b))
  {
    long long n = (long long)B * m;
    int blocks = (int)((n + 255) / 256);
    init_lam_kernel<<<blocks, 256, 0, stream>>>(cMat, b, eta, lamA, m, n);
  }

  // 6) 29 remaining dual PGD steps: lam' = relu(lam + eta*(c - b - lam G))
  float* cur = lamA;
  float* nxt = lamB;
  for (int it = 0; it < 29; ++it) {
    wmma_gemm_nt<1><<<dim3(m / 128, B / 128), 256, 0, stream>>>(
        cur, G, nxt, cMat, b, eta, m, m, m, m);
    float* tmp = cur; cur = nxt; nxt = tmp;
  }

  // 7) out = x - lam A    (Q = AT so the NT core computes lam @ A)
  wmma_gemm_nt<2><<<dim3(d / 128, B / 128), 256, 0, stream>>>(
      cur, AT, out, x, nullptr, nullptr, m, m, m, d);
}